// MatchingLayer_9998683865322
// MI455X (gfx1250) — compile-verified
//
#include <hip/hip_runtime.h>
#include <hip/hip_bf16.h>
#include <math.h>

#define B_  32
#define L_  64
#define H_  200
#define P_  20
#define EPSF 1e-8f
#define LSTR 202           // LDS row stride for 200-float rows (bank-conflict pad)
#define ASTR 66            // LDS row stride for 64-float attention rows
#define CH   160
#define OUTHALF (B_*L_*CH)
#define WSN (B_*L_*H_)     // one derived tensor: 409600 floats

typedef __attribute__((ext_vector_type(2))) float v2f;
typedef __attribute__((ext_vector_type(8))) float v8f;

__device__ __forceinline__ v8f wmma_f32(v2f a, v2f b, v8f c) {
  // D(16x16,f32) = A(16x4,f32) x B(4x16,f32) + C
  return __builtin_amdgcn_wmma_f32_16x16x4_f32(false, a, false, b, (short)0, c,
                                               false, false);
}

// ---------------------------------------------------------------------------
// Kernel 1: multi-perspective pairwise cosine (w2/w3), max over j -> mv_p,
// max over i -> mv_h.  One block per (b, perspective, dir). 4 waves.
// ---------------------------------------------------------------------------
__global__ __launch_bounds__(128) void k_pairwise(
    const float* __restrict__ conP, const float* __restrict__ conH,
    const float* __restrict__ w2, const float* __restrict__ w3,
    float* __restrict__ out)
{
  __shared__ float sP[64*LSTR + 16];
  __shared__ float sH[64*LSTR + 16];
  __shared__ float nP[64], nH[64];
  __shared__ float colpart[4][64];

  const int b = blockIdx.x, pp = blockIdx.y, dir = blockIdx.z;
  const float* w = dir ? w3 : w2;
  const int tid = threadIdx.x;

  // stage w-scaled rows
  for (int idx = tid; idx < 64*H_; idx += 128) {
    int i = idx / H_, h = idx - i*H_;
    float wv = w[pp*H_ + h];
    size_t row = (size_t)(b*64 + i)*400 + (size_t)dir*200;
    sP[i*LSTR + h] = conP[row + h] * wv;
    sH[i*LSTR + h] = conH[row + h] * wv;
  }
  __syncthreads();

  if (tid < 64) {
    float s = 0.f;
    for (int h = 0; h < H_; ++h) { float v = sP[tid*LSTR + h]; s += v*v; }
    nP[tid] = sqrtf(s);
  } else {
    int r = tid - 64;
    float s = 0.f;
    for (int h = 0; h < H_; ++h) { float v = sH[r*LSTR + h]; s += v*v; }
    nH[r] = sqrtf(s);
  }
  __syncthreads();

  const int wave = tid >> 5, lane = tid & 31;
  const int m = lane & 15, hiHalf = lane >> 4;
  const int iBase = wave * 16;

  float rmax[8];
  #pragma unroll
  for (int r = 0; r < 8; ++r) rmax[r] = -3.0e38f;

  for (int jt = 0; jt < 4; ++jt) {
    v8f acc = {};
    const float* arow = &sP[(iBase + m) * LSTR];     // A[m,k] = Pw[iBase+m, k]
    const float* brow = &sH[(jt*16 + m) * LSTR];     // B[k,n] = Hw[jt*16+n, k]
    for (int kk = 0; kk < 50; ++kk) {
      int k = kk*4 + hiHalf*2;
      v2f a  = { arow[k], arow[k+1] };
      v2f bb = { brow[k], brow[k+1] };
      acc = wmma_f32(a, bb, acc);
    }
    float nj = nH[jt*16 + m];
    float cmax = -3.0e38f;
    #pragma unroll
    for (int r = 0; r < 8; ++r) {
      int i = iBase + r + 8*hiHalf;
      float d = nP[i] * nj;
      d = d > EPSF ? d : EPSF;
      float c = acc[r] / d;
      rmax[r] = fmaxf(rmax[r], c);
      cmax    = fmaxf(cmax, c);
    }
    cmax = fmaxf(cmax, __shfl_xor(cmax, 16, 32));    // combine row halves
    if (hiHalf == 0) colpart[wave][jt*16 + m] = cmax;
  }

  // row-max: butterfly within each 16-lane half
  #pragma unroll
  for (int off = 8; off >= 1; off >>= 1) {
    #pragma unroll
    for (int r = 0; r < 8; ++r)
      rmax[r] = fmaxf(rmax[r], __shfl_xor(rmax[r], off, 32));
  }
  const int ch = 40 + 20*dir + pp;
  if (m == 0) {
    #pragma unroll
    for (int r = 0; r < 8; ++r) {
      int i = iBase + r + 8*hiHalf;
      out[(size_t)(b*64 + i)*CH + ch] = rmax[r];
    }
  }
  __syncthreads();
  if (tid < 64) {
    float cm = fmaxf(fmaxf(colpart[0][tid], colpart[1][tid]),
                     fmaxf(colpart[2][tid], colpart[3][tid]));
    out[(size_t)OUTHALF + (size_t)(b*64 + tid)*CH + ch] = cm;
  }
}

// ---------------------------------------------------------------------------
// Kernel 2: attention matrix + derived tensors (att-mean / att-max) into ws.
// One block per (b, dir). 4 waves.
// ws layout (each B*L*H floats): 0/1 att_mean_h fw/bw, 2/3 att_mean_p fw/bw,
//                                4/5 att_h_max fw/bw,  6/7 att_p_max fw/bw
// ---------------------------------------------------------------------------
__global__ __launch_bounds__(128) void k_attention(
    const float* __restrict__ conP, const float* __restrict__ conH,
    float* __restrict__ ws)
{
  __shared__ float sP[64*LSTR + 16];
  __shared__ float sH[64*LSTR + 16];
  __shared__ float att[64*ASTR];
  __shared__ float nP[64], nH[64], rs[64], cs[64];

  const int b = blockIdx.x, dir = blockIdx.y;
  const int tid = threadIdx.x;

  for (int idx = tid; idx < 64*H_; idx += 128) {
    int i = idx / H_, h = idx - i*H_;
    size_t row = (size_t)(b*64 + i)*400 + (size_t)dir*200;
    sP[i*LSTR + h] = conP[row + h];
    sH[i*LSTR + h] = conH[row + h];
  }
  __syncthreads();
  if (tid < 64) {
    float s = 0.f;
    for (int h = 0; h < H_; ++h) { float v = sP[tid*LSTR + h]; s += v*v; }
    nP[tid] = sqrtf(s);
  } else {
    int r = tid - 64;
    float s = 0.f;
    for (int h = 0; h < H_; ++h) { float v = sH[r*LSTR + h]; s += v*v; }
    nH[r] = sqrtf(s);
  }
  __syncthreads();

  const int wave = tid >> 5, lane = tid & 31;
  const int m = lane & 15, hiHalf = lane >> 4;
  const int iBase = wave * 16;

  // cosine attention matrix via WMMA
  for (int jt = 0; jt < 4; ++jt) {
    v8f acc = {};
    const float* arow = &sP[(iBase + m) * LSTR];
    const float* brow = &sH[(jt*16 + m) * LSTR];
    for (int kk = 0; kk < 50; ++kk) {
      int k = kk*4 + hiHalf*2;
      v2f a  = { arow[k], arow[k+1] };
      v2f bb = { brow[k], brow[k+1] };
      acc = wmma_f32(a, bb, acc);
    }
    float nj = nH[jt*16 + m];
    #pragma unroll
    for (int r = 0; r < 8; ++r) {
      int i = iBase + r + 8*hiHalf;
      float d = nP[i] * nj;
      d = d > EPSF ? d : EPSF;
      att[i*ASTR + jt*16 + m] = acc[r] / d;
    }
  }
  __syncthreads();

  if (tid < 64) {
    float s = 0.f;
    for (int j = 0; j < 64; ++j) s += att[tid*ASTR + j];
    rs[tid] = s;
  } else {
    int j = tid - 64;
    float s = 0.f;
    for (int i = 0; i < 64; ++i) s += att[i*ASTR + j];
    cs[j] = s;
  }
  __syncthreads();

  float* amh  = ws + (size_t)(0 + dir)*WSN + (size_t)b*64*H_;
  float* amp  = ws + (size_t)(2 + dir)*WSN + (size_t)b*64*H_;
  float* amxh = ws + (size_t)(4 + dir)*WSN + (size_t)b*64*H_;
  float* amxp = ws + (size_t)(6 + dir)*WSN + (size_t)b*64*H_;

  // att_mean_h = (att @ H) / clamp(rowsum)   -- M=64, K=64, N=200 (13 tiles)
  for (int nt = 0; nt < 13; ++nt) {
    v8f acc = {};
    int ncol = nt*16 + m;
    for (int kk = 0; kk < 16; ++kk) {
      int k = kk*4 + hiHalf*2;
      v2f a  = { att[(iBase + m)*ASTR + k], att[(iBase + m)*ASTR + k + 1] };
      v2f bb = { sH[k*LSTR + ncol],         sH[(k+1)*LSTR + ncol] };
      acc = wmma_f32(a, bb, acc);
    }
    if (ncol < H_) {
      #pragma unroll
      for (int r = 0; r < 8; ++r) {
        int i = iBase + r + 8*hiHalf;
        float d = rs[i]; d = d > EPSF ? d : EPSF;
        amh[(size_t)i*H_ + ncol] = acc[r] / d;
      }
    }
  }
  // att_mean_p = (att^T @ P) / clamp(colsum)
  for (int nt = 0; nt < 13; ++nt) {
    v8f acc = {};
    int ncol = nt*16 + m;
    for (int kk = 0; kk < 16; ++kk) {
      int k = kk*4 + hiHalf*2;
      v2f a  = { att[k*ASTR + iBase + m], att[(k+1)*ASTR + iBase + m] };
      v2f bb = { sP[k*LSTR + ncol],       sP[(k+1)*LSTR + ncol] };
      acc = wmma_f32(a, bb, acc);
    }
    if (ncol < H_) {
      #pragma unroll
      for (int r = 0; r < 8; ++r) {
        int j = iBase + r + 8*hiHalf;
        float d = cs[j]; d = d > EPSF ? d : EPSF;
        amp[(size_t)j*H_ + ncol] = acc[r] / d;
      }
    }
  }

  // elementwise att-weighted maxes
  for (int idx = tid; idx < 64*H_; idx += 128) {
    int i = idx / H_, h = idx - i*H_;
    float mh = -3.0e38f, mp = -3.0e38f;
    for (int j = 0; j < 64; ++j) {
      mh = fmaxf(mh, att[i*ASTR + j] * sH[j*LSTR + h]);  // max_j att[i,j]*h[j,h]
      mp = fmaxf(mp, att[j*ASTR + i] * sP[j*LSTR + h]);  // max_i att[i,(=i)]* -> out row i as "j"
    }
    amxh[(size_t)i*H_ + h] = mh;
    amxp[(size_t)i*H_ + h] = mp;
  }
}

// ---------------------------------------------------------------------------
// Kernel 3: the 12 _mp_match variants (channels 0-39, 80-159 of both halves).
// ---------------------------------------------------------------------------
__global__ __launch_bounds__(256) void k_match(
    const float* __restrict__ conP, const float* __restrict__ conH,
    const float* __restrict__ ws,
    const float* __restrict__ w0, const float* __restrict__ w1,
    const float* __restrict__ w4, const float* __restrict__ w5,
    const float* __restrict__ w6, const float* __restrict__ w7,
    float* __restrict__ out)
{
  const int b = blockIdx.x, c = blockIdx.y;
  const int tid = threadIdx.x;
  const size_t rowB = (size_t)b*64*400;
  const size_t wsb  = (size_t)b*64*H_;

  const float* v1; const float* v2; int v2s; const float* w; int ch; int side;
  switch (c) {
    case 0:  v1 = conP + rowB;       v2 = conH + rowB + 63*400;     v2s = 0;   w = w0; ch = 0;   side = 0; break;
    case 1:  v1 = conP + rowB + 200; v2 = conH + rowB + 200;        v2s = 0;   w = w1; ch = 20;  side = 0; break;
    case 2:  v1 = conP + rowB;       v2 = ws + 0*(size_t)WSN + wsb; v2s = H_;  w = w4; ch = 80;  side = 0; break;
    case 3:  v1 = conP + rowB + 200; v2 = ws + 1*(size_t)WSN + wsb; v2s = H_;  w = w5; ch = 100; side = 0; break;
    case 4:  v1 = conP + rowB;       v2 = ws + 4*(size_t)WSN + wsb; v2s = H_;  w = w6; ch = 120; side = 0; break;
    case 5:  v1 = conP + rowB + 200; v2 = ws + 5*(size_t)WSN + wsb; v2s = H_;  w = w7; ch = 140; side = 0; break;
    case 6:  v1 = conH + rowB;       v2 = conP + rowB + 63*400;     v2s = 0;   w = w0; ch = 0;   side = 1; break;
    case 7:  v1 = conH + rowB + 200; v2 = conP + rowB + 200;        v2s = 0;   w = w1; ch = 20;  side = 1; break;
    case 8:  v1 = conH + rowB;       v2 = ws + 2*(size_t)WSN + wsb; v2s = H_;  w = w4; ch = 80;  side = 1; break;
    case 9:  v1 = conH + rowB + 200; v2 = ws + 3*(size_t)WSN + wsb; v2s = H_;  w = w5; ch = 100; side = 1; break;
    case 10: v1 = conH + rowB;       v2 = ws + 6*(size_t)WSN + wsb; v2s = H_;  w = w6; ch = 120; side = 1; break;
    default: v1 = conH + rowB + 200; v2 = ws + 7*(size_t)WSN + wsb; v2s = H_;  w = w7; ch = 140; side = 1; break;
  }

  for (int o = tid; o < 64*P_; o += 256) {
    int l = o / P_, q = o - l*P_;
    const float* a  = v1 + (size_t)l*400;
    const float* v  = v2 + (size_t)l*v2s;
    const float* wr = w + q*H_;
    float num = 0.f, s1 = 0.f, s2 = 0.f;
    for (int h = 0; h < H_; ++h) {
      float wv = wr[h]; float ww = wv*wv;
      float x = a[h], y = v[h];
      num += x*y*ww; s1 += x*x*ww; s2 += y*y*ww;
    }
    float d = sqrtf(s1) * sqrtf(s2);
    d = d > EPSF ? d : EPSF;
    out[(size_t)side*OUTHALF + (size_t)(b*64 + l)*CH + ch + q] = num / d;
  }
}

extern "C" void kernel_launch(void* const* d_in, const int* in_sizes, int n_in,
                              void* d_out, int out_size, void* d_ws, size_t ws_size,
                              hipStream_t stream) {
  const float* conP = (const float*)d_in[0];
  const float* conH = (const float*)d_in[1];
  const float* w0 = (const float*)d_in[2];
  const float* w1 = (const float*)d_in[3];
  const float* w2 = (const float*)d_in[4];
  const float* w3 = (const float*)d_in[5];
  const float* w4 = (const float*)d_in[6];
  const float* w5 = (const float*)d_in[7];
  const float* w6 = (const float*)d_in[8];
  const float* w7 = (const float*)d_in[9];
  float* out = (float*)d_out;
  float* ws  = (float*)d_ws;   // needs 8 * 32*64*200 * 4 = 13.1 MB

  k_pairwise <<<dim3(B_, P_, 2), 128, 0, stream>>>(conP, conH, w2, w3, out);
  k_attention<<<dim3(B_, 2),     128, 0, stream>>>(conP, conH, ws);
  k_match    <<<dim3(B_, 12),    256, 0, stream>>>(conP, conH, ws,
                                                   w0, w1, w4, w5, w6, w7, out);
}